// DepthPolarReducer_69973607186476
// MI455X (gfx1250) — compile-verified
//
#include <hip/hip_runtime.h>
#include <hip/hip_bf16.h>
#include <math.h>

// ---- problem constants (from reference) ----
#define W_IMG      640
#define H_IMG      480
#define CROP_START 192
#define CROP_H     288
#define BATCH      64
#define NUM_BINS   32
#define KAPPA      20.0f
#define SPLIT_H    4
#define ROWS_PER   (CROP_H / SPLIT_H)   // 72

typedef __attribute__((ext_vector_type(2))) float v2f;
typedef __attribute__((ext_vector_type(8))) float v8f;

// ------------------------------------------------------------------
// k1: streaming online log-sum-exp partials over a slab of 72 rows.
//   pm[part][b][x] = running max of neg = -20*dm
//   ps[part][b][x] = sum of mask * exp(neg - pm)
// 320 threads, each owns 2 adjacent columns (float2 loads, coalesced).
// ------------------------------------------------------------------
__global__ __launch_bounds__(320) void k1_partial(
    const float* __restrict__ depth, const float* __restrict__ mask,
    float* __restrict__ pm, float* __restrict__ ps) {
  const int part = blockIdx.x;        // 0..SPLIT_H-1
  const int b    = blockIdx.y;        // 0..63
  const int x0   = threadIdx.x * 2;   // 0..638
  const int h0   = CROP_START + part * ROWS_PER;
  const float2* dp = (const float2*)(depth + ((size_t)b * H_IMG + h0) * W_IMG + x0);
  const float2* mp = (const float2*)(mask  + ((size_t)b * H_IMG + h0) * W_IMG + x0);

  float m0 = -INFINITY, m1 = -INFINITY, s0 = 0.f, s1 = 0.f;
  for (int h = 0; h < ROWS_PER; ++h) {
    if (h + 16 < ROWS_PER) {  // keep the HBM pipe full (global_prefetch)
      __builtin_prefetch((const char*)(dp + (size_t)(h + 16) * (W_IMG / 2)), 0, 1);
      __builtin_prefetch((const char*)(mp + (size_t)(h + 16) * (W_IMG / 2)), 0, 1);
    }
    float2 d  = dp[(size_t)h * (W_IMG / 2)];
    float2 mk = mp[(size_t)h * (W_IMG / 2)];
    float dm0 = d.x * mk.x + (1.f - mk.x) * 100.f;
    float dm1 = d.y * mk.y + (1.f - mk.y) * 100.f;
    float n0 = -KAPPA * dm0, n1 = -KAPPA * dm1;
    float t0 = fmaxf(m0, n0), t1 = fmaxf(m1, n1);
    s0 = s0 * __expf(m0 - t0) + mk.x * __expf(n0 - t0);
    s1 = s1 * __expf(m1 - t1) + mk.y * __expf(n1 - t1);
    m0 = t0; m1 = t1;
  }
  size_t o = ((size_t)part * BATCH + b) * W_IMG + x0;
  pm[o] = m0; pm[o + 1] = m1;
  ps[o] = s0; ps[o + 1] = s1;
}

// ------------------------------------------------------------------
// k2: merge SPLIT_H partials -> colmax[b,x], col; block-reduce
//     Cb[b] = max_x colmax; emit P[b,x] = exp(colmax - Cb) * col.
// One block per batch, 320 threads x 2 columns.
// ------------------------------------------------------------------
__global__ __launch_bounds__(320) void k2_merge(
    const float* __restrict__ pm, const float* __restrict__ ps,
    float* __restrict__ colmax, float* __restrict__ P, float* __restrict__ Cb) {
  const int b = blockIdx.x;
  const int t = threadIdx.x;
  float cm[2], cl[2];
#pragma unroll
  for (int j = 0; j < 2; ++j) {
    const int x = t * 2 + j;
    float m = -INFINITY;
#pragma unroll
    for (int p = 0; p < SPLIT_H; ++p)
      m = fmaxf(m, pm[((size_t)p * BATCH + b) * W_IMG + x]);
    float s = 0.f;
#pragma unroll
    for (int p = 0; p < SPLIT_H; ++p) {
      size_t o = ((size_t)p * BATCH + b) * W_IMG + x;
      s += ps[o] * __expf(pm[o] - m);
    }
    cm[j] = m; cl[j] = s;
    colmax[(size_t)b * W_IMG + x] = m;
  }
  __shared__ float red[512];
  red[t] = fmaxf(cm[0], cm[1]);
  if (t < 192) red[t + 320] = -INFINITY;
  __syncthreads();
  for (int st = 256; st > 0; st >>= 1) {
    if (t < st) red[t] = fmaxf(red[t], red[t + st]);
    __syncthreads();
  }
  const float cb = red[0];
  if (t == 0) Cb[b] = cb;
#pragma unroll
  for (int j = 0; j < 2; ++j) {
    const int x = t * 2 + j;
    P[(size_t)b * W_IMG + x] = __expf(cm[j] - cb) * cl[j];
  }
}

// ------------------------------------------------------------------
// k3: exact M[b,bin] = max_x (colmax[b,x] + log(w[bin,x]+1e-10)).
// One block per batch; blockDim (32,32): wave = one bin, 32 x-lanes.
// ------------------------------------------------------------------
__global__ __launch_bounds__(1024) void k3_M(
    const float* __restrict__ colmax, const float* __restrict__ w,
    float* __restrict__ M) {
  const int b   = blockIdx.x;
  const int lx  = threadIdx.x;   // 0..31 (lane)
  const int bin = threadIdx.y;   // 0..31
  float acc = -INFINITY;
#pragma unroll
  for (int j = 0; j < W_IMG / 32; ++j) {
    const int x = lx + 32 * j;
    float lw = __logf(w[(size_t)bin * W_IMG + x] + 1e-10f);
    acc = fmaxf(acc, colmax[(size_t)b * W_IMG + x] + lw);
  }
#pragma unroll
  for (int st = 16; st > 0; st >>= 1)
    acc = fmaxf(acc, __shfl_xor(acc, st, 32));
  if (lx == 0) M[(size_t)b * NUM_BINS + bin] = acc;
}

// ------------------------------------------------------------------
// k4: G[b,bin] = sum_x P[b,x] * (w[bin,x]+1e-10) as an f32 WMMA GEMM
// (64x640)*(640x32) with V_WMMA_F32_16X16X4_F32, then the final
// S = exp(Cb-M)*G + 1e-10, clearance = -(M + log S)/KAPPA, w_sum mask.
// 8 waves (1 block of 256): wave -> one 16x16 C tile; K-loop of 160.
// A 16x4 f32 layout: lane l holds row (l&15); K = 2*(l>>4)+{0,1}.
// B 4x16 f32 layout: lane l holds col (l&15); K = 2*(l>>4)+{0,1}.
// C layout: lane l, VGPR v -> row v + 8*(l>>4), col (l&15).
// ------------------------------------------------------------------
__global__ __launch_bounds__(256) void k4_wmma(
    const float* __restrict__ P, const float* __restrict__ w,
    const float* __restrict__ Cb, const float* __restrict__ M,
    float* __restrict__ out) {
  const int wave = threadIdx.x >> 5;   // 0..7
  const int lane = threadIdx.x & 31;
  const int bt = wave >> 1;            // batch tile 0..3
  const int nt = wave & 1;             // bin tile 0..1
  const int b0 = bt * 16, n0 = nt * 16;
  const int r    = lane & 15;
  const int half = lane >> 4;
  const int bin  = n0 + r;             // this lane's output column

  v8f c = {0.f, 0.f, 0.f, 0.f, 0.f, 0.f, 0.f, 0.f};
  float wsum_part = 0.f;
  for (int xc = 0; xc < W_IMG; xc += 4) {
    const int xa = xc + 2 * half;
    v2f a, bb;
    a.x = P[(size_t)(b0 + r) * W_IMG + xa];
    a.y = P[(size_t)(b0 + r) * W_IMG + xa + 1];
    float w0 = w[(size_t)bin * W_IMG + xa];
    float w1 = w[(size_t)bin * W_IMG + xa + 1];
    wsum_part += w0 + w1;              // raw w for the w_sum mask
    bb.x = w0 + 1e-10f;                // floored weight for the GEMM
    bb.y = w1 + 1e-10f;
    c = __builtin_amdgcn_wmma_f32_16x16x4_f32(
        false, a, false, bb, (short)0, c, false, false);
  }
  // lanes l and l^16 covered complementary x mod 4 -> combine
  const float wsum = (wsum_part + __shfl_xor(wsum_part, 16, 32)) * (float)CROP_H;

#pragma unroll
  for (int v = 0; v < 8; ++v) {
    const int b  = b0 + v + 8 * half;
    const float Mv = M[(size_t)b * NUM_BINS + bin];
    const float S  = __expf(Cb[b] - Mv) * c[v] + 1e-10f;
    const float clr = -(Mv + __logf(S)) * (1.0f / KAPPA);
    out[(size_t)b * NUM_BINS + bin] = (wsum < 1e-6f) ? 100.f : clr;
  }
}

// ------------------------------------------------------------------
extern "C" void kernel_launch(void* const* d_in, const int* in_sizes, int n_in,
                              void* d_out, int out_size, void* d_ws, size_t ws_size,
                              hipStream_t stream) {
  const float* depth = (const float*)d_in[0];   // (64,480,640) f32
  const float* mask  = (const float*)d_in[1];   // (64,480,640) f32
  const float* binw  = (const float*)d_in[2];   // (32,640) f32
  float* out = (float*)d_out;                   // (64,32) f32

  // workspace layout (floats): ~1.65 MB total
  float* ws     = (float*)d_ws;
  float* pm     = ws;                                       // 4*64*640
  float* ps     = pm + (size_t)SPLIT_H * BATCH * W_IMG;     // 4*64*640
  float* colmax = ps + (size_t)SPLIT_H * BATCH * W_IMG;     // 64*640
  float* P      = colmax + (size_t)BATCH * W_IMG;           // 64*640
  float* Cb     = P + (size_t)BATCH * W_IMG;                // 64
  float* M      = Cb + BATCH;                               // 64*32

  k1_partial<<<dim3(SPLIT_H, BATCH), 320, 0, stream>>>(depth, mask, pm, ps);
  k2_merge<<<dim3(BATCH), 320, 0, stream>>>(pm, ps, colmax, P, Cb);
  k3_M<<<dim3(BATCH), dim3(32, 32), 0, stream>>>(colmax, binw, M);
  k4_wmma<<<dim3(1), 256, 0, stream>>>(P, binw, Cb, M, out);
}